// Env_61744449848046
// MI455X (gfx1250) — compile-verified
//
#include <hip/hip_runtime.h>
#include <hip/hip_bf16.h>

typedef __attribute__((ext_vector_type(2))) float v2f;
typedef __attribute__((ext_vector_type(8))) float v8f;

#define FDIM 128
#define PDIM 16

// ---------------------------------------------------------------------------
// Zero-fill the 256MB output with wide stores (HBM-bound: ~11us floor).
// ---------------------------------------------------------------------------
__global__ __launch_bounds__(256) void zero_kernel(float4* __restrict__ out, size_t n4) {
    size_t i = (size_t)blockIdx.x * blockDim.x + threadIdx.x;
    size_t stride = (size_t)gridDim.x * blockDim.x;
    float4 z = make_float4(0.f, 0.f, 0.f, 0.f);
    for (; i < n4; i += stride) out[i] = z;
}

// ---------------------------------------------------------------------------
// One wave32 per row: diff = feature - next_feature, nf = double-L2-normalize
// (next_feature), and persona-weighted alpha/beta/gamma row scalars.
// Each lane owns 4 consecutive features (float4 = global_load_b128).
// ---------------------------------------------------------------------------
__device__ __forceinline__ float wave_reduce_sum(float v) {
#pragma unroll
    for (int off = 16; off > 0; off >>= 1) v += __shfl_xor(v, off, 32);
    return v;
}

__global__ __launch_bounds__(256) void row_prep_kernel(
    const float* __restrict__ feature, const float* __restrict__ next_feature,
    const float* __restrict__ persona_t,
    const float* __restrict__ alpha, const float* __restrict__ beta,
    const float* __restrict__ gamma,
    float* __restrict__ diff, float* __restrict__ nf,
    float* __restrict__ alpha_row, float* __restrict__ beta_row,
    float* __restrict__ gamma_row, int n)
{
    const int lane = threadIdx.x & 31;
    const int row  = ((int)blockIdx.x * (int)blockDim.x + (int)threadIdx.x) >> 5;
    if (row >= n) return;
    const size_t rowOff = (size_t)row * FDIM;

    float4 fv = ((const float4*)(feature + rowOff))[lane];
    float4 nv = ((const float4*)(next_feature + rowOff))[lane];

    float4 dv = make_float4(fv.x - nv.x, fv.y - nv.y, fv.z - nv.z, fv.w - nv.w);
    ((float4*)(diff + rowOff))[lane] = dv;

    // first normalize
    float ss = wave_reduce_sum(nv.x*nv.x + nv.y*nv.y + nv.z*nv.z + nv.w*nv.w);
    float inv = 1.0f / sqrtf(ss);
    float4 y = make_float4(nv.x*inv, nv.y*inv, nv.z*inv, nv.w*inv);
    // second normalize (reference normalizes twice)
    float ss2 = wave_reduce_sum(y.x*y.x + y.y*y.y + y.z*y.z + y.w*y.w);
    float inv2 = 1.0f / sqrtf(ss2);
    ((float4*)(nf + rowOff))[lane] =
        make_float4(y.x*inv2, y.y*inv2, y.z*inv2, y.w*inv2);

    // persona dots: lanes 0..15 hold the 16 persona components
    float p = (lane < PDIM) ? persona_t[(size_t)row * PDIM + lane] : 0.f;
    float av = (lane < PDIM) ? alpha[lane] : 0.f;
    float bv = (lane < PDIM) ? beta[lane]  : 0.f;
    float gv = (lane < PDIM) ? gamma[lane] : 0.f;
    float sa = wave_reduce_sum(p * av);
    float sb = wave_reduce_sum(p * bv);
    float sg = wave_reduce_sum(p * gv);
    if (lane == 0) {
        alpha_row[row] = sa;
        beta_row[row]  = sb;
        gamma_row[row] = sg;
    }
}

// ---------------------------------------------------------------------------
// 16 edges per wave. Two f32 WMMA accumulation chains (diff path and nf path):
//   D = A(16x4) x B(4x16) + C, chained 32x over K=128.
// Per-edge dot = diagonal of the 16x16 result.
// A layout: lanes 0-15 hold {K=0,K=1} in VGPR{0,1}; lanes 16-31 hold {K=2,K=3}.
// B layout (4x16, row-striped across lanes): VGPR0 = row K=0|K=2, VGPR1 = K=1|K=3.
// Both reduce to the same per-lane addressing: (row, k0+ksub .. k0+ksub+1).
// Diagonal [i][i]: lanes 0-7 hold diag[0..7] in c[lane]; lanes 24-31 hold
// diag[8..15] in c[lane-24].
// ---------------------------------------------------------------------------
__global__ __launch_bounds__(256) void edge_wmma_kernel(
    const float* __restrict__ diff, const float* __restrict__ nf,
    const float* __restrict__ alpha_row, const float* __restrict__ gamma_row,
    const int* __restrict__ act_src, const int* __restrict__ act_dst,
    float* __restrict__ out, int n, int nedges)
{
    const int lane  = (int)threadIdx.x & 31;
    const int wave  = ((int)blockIdx.x * (int)blockDim.x + (int)threadIdx.x) >> 5;
    const int base  = wave * 16;
    if (base >= nedges) return;    // full waves quit together; EXEC stays all-1

    const int m    = lane & 15;            // which of the 16 edges this lane feeds
    const int ksub = (lane >> 4) << 1;     // 0 for lanes 0-15, 2 for lanes 16-31

    int e = base + m;
    if (e >= nedges) e = nedges - 1;       // clamp (E is a multiple of 16 anyway)
    const int s = act_src[e];
    const int d = act_dst[e];
    const float* __restrict__ srcDiff = diff + (size_t)s * FDIM;
    const float* __restrict__ dstDiff = diff + (size_t)d * FDIM;
    const float* __restrict__ srcNf   = nf   + (size_t)s * FDIM;
    const float* __restrict__ dstNf   = nf   + (size_t)d * FDIM;

    v8f c_imp = {};
    v8f c_sim = {};
#pragma unroll
    for (int k0 = 0; k0 < FDIM; k0 += 4) {
        const int k = k0 + ksub;
        v2f a_imp = { srcDiff[k], srcDiff[k + 1] };
        v2f b_imp = { dstDiff[k], dstDiff[k + 1] };
        v2f a_sim = { srcNf[k],   srcNf[k + 1]   };
        v2f b_sim = { dstNf[k],   dstNf[k + 1]   };
        c_imp = __builtin_amdgcn_wmma_f32_16x16x4_f32(
            false, a_imp, false, b_imp, (short)0, c_imp, false, false);
        c_sim = __builtin_amdgcn_wmma_f32_16x16x4_f32(
            false, a_sim, false, b_sim, (short)0, c_sim, false, false);
    }

    // Extract the diagonal from the 16x16 f32 C/D layout.
    const bool active = (lane < 8) || (lane >= 24);
    const int  r = (lane < 8) ? lane : (lane - 24);   // VGPR index holding diag
    float dImp = 0.f, dSim = 0.f;
#pragma unroll
    for (int j = 0; j < 8; ++j) {
        if (j == r) { dImp = c_imp[j]; dSim = c_sim[j]; }
    }

    if (active) {
        const int i  = (lane < 8) ? lane : (lane - 16);  // edge within group
        const int ee = base + i;
        if (ee < nedges) {
            const int ss = act_src[ee];
            const int dd = act_dst[ee];
            const float val = dSim * alpha_row[ss]
                            + dImp * (1.0f / (float)FDIM) * gamma_row[ss];
            atomicAdd(out + (size_t)ss * (size_t)n + (size_t)dd, val);
        }
    }
}

// ---------------------------------------------------------------------------
// Cost path: out[edge_src, edge_dst] -= beta_row[edge_src]
// ---------------------------------------------------------------------------
__global__ __launch_bounds__(256) void cost_kernel(
    const int* __restrict__ edge_src, const int* __restrict__ edge_dst,
    const float* __restrict__ beta_row, float* __restrict__ out,
    int n, int nedges)
{
    const int e = (int)blockIdx.x * (int)blockDim.x + (int)threadIdx.x;
    if (e < nedges) {
        const int s = edge_src[e];
        const int d = edge_dst[e];
        atomicAdd(out + (size_t)s * (size_t)n + (size_t)d, -beta_row[s]);
    }
}

extern "C" void kernel_launch(void* const* d_in, const int* in_sizes, int n_in,
                              void* d_out, int out_size, void* d_ws, size_t ws_size,
                              hipStream_t stream) {
    const float* feature      = (const float*)d_in[0];
    const float* next_feature = (const float*)d_in[1];
    const float* persona_t    = (const float*)d_in[2];
    const float* alpha        = (const float*)d_in[3];
    const float* beta         = (const float*)d_in[4];
    const float* gamma        = (const float*)d_in[5];
    const int*   act_src      = (const int*)d_in[6];
    const int*   act_dst      = (const int*)d_in[7];
    const int*   edge_src     = (const int*)d_in[8];
    const int*   edge_dst     = (const int*)d_in[9];
    float*       out          = (float*)d_out;

    const int n = in_sizes[0] / FDIM;     // 8192
    const int e = in_sizes[6];            // 262144

    // Workspace layout (floats)
    float* ws        = (float*)d_ws;
    float* diff      = ws;                                  // n*FDIM
    float* nf        = diff + (size_t)n * FDIM;             // n*FDIM
    float* alpha_row = nf   + (size_t)n * FDIM;             // n
    float* gamma_row = alpha_row + n;                       // n
    float* beta_row  = gamma_row + n;                       // n

    // 1) zero output (256MB, float4 stores)
    const size_t n4 = ((size_t)out_size) / 4;
    zero_kernel<<<4096, 256, 0, stream>>>((float4*)out, n4);

    // 2) per-row prep: one wave per row
    {
        const int waves = n;
        const int blocks = (waves * 32 + 255) / 256;
        row_prep_kernel<<<blocks, 256, 0, stream>>>(
            feature, next_feature, persona_t, alpha, beta, gamma,
            diff, nf, alpha_row, beta_row, gamma_row, n);
    }

    // 3) WMMA edge dot products + scatter (16 edges per wave)
    {
        const int waves = (e + 15) / 16;
        const int blocks = (waves * 32 + 255) / 256;
        edge_wmma_kernel<<<blocks, 256, 0, stream>>>(
            diff, nf, alpha_row, gamma_row, act_src, act_dst, out, n, e);
    }

    // 4) cost edges
    {
        const int blocks = (e + 255) / 256;
        cost_kernel<<<blocks, 256, 0, stream>>>(
            edge_src, edge_dst, beta_row, out, n, e);
    }
}